// Graph_GAN_66872640798821
// MI455X (gfx1250) — compile-verified
//
#include <hip/hip_runtime.h>
#include <hip/hip_bf16.h>
#include <cstdint>

// Problem constants (match reference)
#define B_   32
#define N_   128
#define H_   32
#define E0   96     // edge MLP layer widths
#define E1   160
#define E2   192
#define F0   224    // node MLP widths (192 + 32)
#define F1   256
#define F2   256
#define F3   32
#define ALPHA_ 0.2f
#define NODES (B_ * N_)   // 4096

typedef __attribute__((ext_vector_type(2))) float v2f;
typedef __attribute__((ext_vector_type(8))) float v8f;

// V_WMMA_F32_16X16X4_F32 : D(16x16 f32) = A(16x4) * B(4x16) + C
// 8-arg pattern: (neg_a, A, neg_b, B, c_mod, C, reuse_a, reuse_b)
__device__ __forceinline__ v8f wmma4(v2f a, v2f b, v8f c) {
  return __builtin_amdgcn_wmma_f32_16x16x4_f32(
      false, a, false, b, (short)0, c, false, false);
}

__device__ __forceinline__ float lrelu(float v) { return v > 0.f ? v : ALPHA_ * v; }

// ---------------------------------------------------------------------------
// P[n] = x[n] @ W0[0:32,:] + b0   (96)    -- i-side of edge layer 0
// Q[n] = x[n] @ W0[32:64,:]       (96)    -- j-side of edge layer 0
// ---------------------------------------------------------------------------
__global__ __launch_bounds__(192) void pq_kernel(
    const float* __restrict__ x, const float* __restrict__ W0,
    const float* __restrict__ b0, float* __restrict__ P, float* __restrict__ Q) {
  __shared__ float sx[H_];
  const int node = blockIdx.x;
  const int tid  = threadIdx.x;
  if (tid < H_) sx[tid] = x[node * H_ + tid];
  __syncthreads();
  if (tid < E0) {
    float s = b0[tid];
    #pragma unroll
    for (int k = 0; k < H_; ++k) s += sx[k] * W0[k * E0 + tid];
    P[(size_t)node * E0 + tid] = s;
  } else {
    const int c = tid - E0;
    float s = 0.f;
    #pragma unroll
    for (int k = 0; k < H_; ++k) s += sx[k] * W0[(H_ + k) * E0 + c];
    Q[(size_t)node * E0 + c] = s;
  }
}

// ---------------------------------------------------------------------------
// Fused edge MLP + neighbor reduction. One workgroup per (b, i) node:
//   A0[j,:] = lrelu(P[i] + Q[j] + d_ij * W0row64)        (128 x 96, LDS)
//   A1      = lrelu(A0 @ W1 + b1)                        (128 x 160, LDS)
//   agg[i]  = sum_j lrelu(A1 @ W2 + b2)[j,:]             (192, registers)
// 10 waves; each wave owns 16-wide output column strips; B fragments are
// register-resident and reused across the 8 row tiles.
// ---------------------------------------------------------------------------
__global__ __launch_bounds__(320) void edge_kernel(
    const float* __restrict__ x,     // [4096][32]
    const float* __restrict__ P,     // [4096][96]  (includes b0)
    const float* __restrict__ Q,     // [4096][96]
    const float* __restrict__ w64,   // [96] = W0 row 64 (dist row)
    const float* __restrict__ W1, const float* __restrict__ b1,   // [96][160],[160]
    const float* __restrict__ W2, const float* __restrict__ b2,   // [160][192],[192]
    float* __restrict__ agg)         // [4096][192]
{
  __shared__ float sA0[N_ * E0];   // 48 KB
  __shared__ float sA1[N_ * E1];   // 80 KB
  __shared__ float sd[N_];
  __shared__ float sPi[E0];
  __shared__ float sw[E0];

  const int bi   = blockIdx.x;      // b*128 + i
  const int b    = bi >> 7;
  const int tid  = threadIdx.x;
  const int lane = tid & 31;
  const int wave = tid >> 5;        // 0..9
  const int kh   = lane >> 4;       // lane half (K/M-half select)
  const int l16  = lane & 15;

  // distances d_ij = ||x[b,j,0:3] - x[b,i,0:3] + 1e-12||
  const float* xb = x + (size_t)b * N_ * H_;
  const float* xi = x + (size_t)bi * H_;
  const float xi0 = xi[0], xi1 = xi[1], xi2 = xi[2];
  for (int j = tid; j < N_; j += 320) {
    const float* xj = xb + j * H_;
    float d0 = xj[0] - xi0 + 1e-12f;
    float d1 = xj[1] - xi1 + 1e-12f;
    float d2 = xj[2] - xi2 + 1e-12f;
    sd[j] = sqrtf(d0 * d0 + d1 * d1 + d2 * d2);
  }
  for (int c = tid; c < E0; c += 320) {
    sPi[c] = P[(size_t)bi * E0 + c];
    sw[c]  = w64[c];
  }
  __syncthreads();

  // ---- layer 0 (factored) ----
  for (int e = tid; e < N_ * E0; e += 320) {
    const int j = e / E0, c = e - j * E0;
    const float v = sPi[c] + Q[((size_t)b * N_ + j) * E0 + c] + sd[j] * sw[c];
    sA0[e] = lrelu(v);
  }
  __syncthreads();

  // ---- layer 1: 128x96 @ 96x160 -> sA1 ----
  {
    const int cs = wave;                 // 10 strips over 10 waves
    const int n0 = cs * 16;
    v2f bf[24];
    #pragma unroll
    for (int kk = 0; kk < 24; ++kk) {
      const int k = kk * 4 + kh * 2;
      bf[kk].x = W1[(k)     * E1 + n0 + l16];
      bf[kk].y = W1[(k + 1) * E1 + n0 + l16];
    }
    const float bias = b1[n0 + l16];
    for (int rt = 0; rt < 8; ++rt) {
      const int r0 = rt * 16;
      v8f acc = {};
      #pragma unroll
      for (int kk = 0; kk < 24; ++kk) {
        const int k = kk * 4 + kh * 2;
        const float* ap = &sA0[(r0 + l16) * E0 + k];
        v2f af; af.x = ap[0]; af.y = ap[1];
        acc = wmma4(af, bf[kk], acc);
      }
      #pragma unroll
      for (int m = 0; m < 8; ++m) {
        const int row = r0 + kh * 8 + m;
        sA1[row * E1 + n0 + l16] = lrelu(acc[m] + bias);
      }
    }
  }
  __syncthreads();

  // ---- layer 2 + sum over j: 128x160 @ 160x192, reduce rows ----
  for (int cs = wave; cs < E2 / 16; cs += 10) {   // 12 strips over 10 waves
    const int n0 = cs * 16;
    v2f bf[40];
    #pragma unroll
    for (int kk = 0; kk < 40; ++kk) {
      const int k = kk * 4 + kh * 2;
      bf[kk].x = W2[(k)     * E2 + n0 + l16];
      bf[kk].y = W2[(k + 1) * E2 + n0 + l16];
    }
    const float bias = b2[n0 + l16];
    float colsum = 0.f;
    for (int rt = 0; rt < 8; ++rt) {
      const int r0 = rt * 16;
      v8f acc = {};
      #pragma unroll
      for (int kk = 0; kk < 40; ++kk) {
        const int k = kk * 4 + kh * 2;
        const float* ap = &sA1[(r0 + l16) * E1 + k];
        v2f af; af.x = ap[0]; af.y = ap[1];
        acc = wmma4(af, bf[kk], acc);
      }
      #pragma unroll
      for (int m = 0; m < 8; ++m) colsum += lrelu(acc[m] + bias);
    }
    colsum += __shfl_xor(colsum, 16, 32);    // combine M-halves
    if (kh == 0) agg[(size_t)bi * E2 + n0 + l16] = colsum;
  }
}

// ---------------------------------------------------------------------------
// Fused node MLP: h = concat(agg, x) : 224 -> 256 -> 256 -> 32.
// 16 rows per workgroup, 8 waves.
// ---------------------------------------------------------------------------
__global__ __launch_bounds__(256) void node_kernel(
    const float* __restrict__ agg, const float* __restrict__ xin,
    const float* __restrict__ Wa, const float* __restrict__ ba,   // 224x256
    const float* __restrict__ Wb, const float* __restrict__ bb,   // 256x256
    const float* __restrict__ Wc, const float* __restrict__ bc,   // 256x32
    float* __restrict__ xout)                                     // [4096][32]
{
  __shared__ float sIn[16 * F0];
  __shared__ float sH1[16 * F1];
  __shared__ float sH2[16 * F2];

  const int r0g  = blockIdx.x * 16;
  const int tid  = threadIdx.x;
  const int lane = tid & 31;
  const int wave = tid >> 5;    // 0..7
  const int kh   = lane >> 4;
  const int l16  = lane & 15;

  for (int e = tid; e < 16 * F0; e += 256) {
    const int r = e / F0, c = e - r * F0;
    sIn[e] = (c < E2) ? agg[(size_t)(r0g + r) * E2 + c]
                      : xin[(size_t)(r0g + r) * H_ + (c - E2)];
  }
  __syncthreads();

  // layer a: 224 -> 256
  for (int cs = wave; cs < 16; cs += 8) {
    const int n0 = cs * 16;
    v8f acc = {};
    #pragma unroll
    for (int kk = 0; kk < F0 / 4; ++kk) {
      const int k = kk * 4 + kh * 2;
      v2f af; af.x = sIn[l16 * F0 + k]; af.y = sIn[l16 * F0 + k + 1];
      v2f bf; bf.x = Wa[(k) * F1 + n0 + l16]; bf.y = Wa[(k + 1) * F1 + n0 + l16];
      acc = wmma4(af, bf, acc);
    }
    const float bias = ba[n0 + l16];
    #pragma unroll
    for (int m = 0; m < 8; ++m)
      sH1[(kh * 8 + m) * F1 + n0 + l16] = lrelu(acc[m] + bias);
  }
  __syncthreads();

  // layer b: 256 -> 256
  for (int cs = wave; cs < 16; cs += 8) {
    const int n0 = cs * 16;
    v8f acc = {};
    #pragma unroll
    for (int kk = 0; kk < F1 / 4; ++kk) {
      const int k = kk * 4 + kh * 2;
      v2f af; af.x = sH1[l16 * F1 + k]; af.y = sH1[l16 * F1 + k + 1];
      v2f bf; bf.x = Wb[(k) * F2 + n0 + l16]; bf.y = Wb[(k + 1) * F2 + n0 + l16];
      acc = wmma4(af, bf, acc);
    }
    const float bias = bb[n0 + l16];
    #pragma unroll
    for (int m = 0; m < 8; ++m)
      sH2[(kh * 8 + m) * F2 + n0 + l16] = lrelu(acc[m] + bias);
  }
  __syncthreads();

  // layer c: 256 -> 32 (linear), waves 0..1 only (wave-uniform branch)
  if (wave < 2) {
    const int n0 = wave * 16;
    v8f acc = {};
    #pragma unroll
    for (int kk = 0; kk < F2 / 4; ++kk) {
      const int k = kk * 4 + kh * 2;
      v2f af; af.x = sH2[l16 * F2 + k]; af.y = sH2[l16 * F2 + k + 1];
      v2f bf; bf.x = Wc[(k) * F3 + n0 + l16]; bf.y = Wc[(k + 1) * F3 + n0 + l16];
      acc = wmma4(af, bf, acc);
    }
    const float bias = bc[n0 + l16];
    #pragma unroll
    for (int m = 0; m < 8; ++m)
      xout[(size_t)(r0g + kh * 8 + m) * F3 + n0 + l16] = acc[m] + bias;
  }
}

__global__ __launch_bounds__(256) void tanh_kernel(const float* __restrict__ xin,
                                                   float* __restrict__ out) {
  const int idx = blockIdx.x * 256 + threadIdx.x;  // 0 .. 12287
  if (idx < NODES * 3) {
    const int node = idx / 3, c = idx - node * 3;
    out[idx] = tanhf(xin[(size_t)node * H_ + c]);
  }
}

// ---------------------------------------------------------------------------
extern "C" void kernel_launch(void* const* d_in, const int* in_sizes, int n_in,
                              void* d_out, int out_size, void* d_ws, size_t ws_size,
                              hipStream_t stream) {
  (void)in_sizes; (void)n_in; (void)out_size; (void)ws_size;

  // workspace layout (floats): P | Q | agg | xbuf0 | xbuf1   (~7 MB)
  float* ws  = (float*)d_ws;
  float* P   = ws;
  float* Q   = P   + (size_t)NODES * E0;
  float* agg = Q   + (size_t)NODES * E0;
  float* xb0 = agg + (size_t)NODES * E2;
  float* xb1 = xb0 + (size_t)NODES * H_;

  const float* xcur  = (const float*)d_in[0];
  float*       xnext = xb0;

  for (int s = 0; s < 2; ++s) {
    const float* W0 = (const float*)d_in[1 + 6 * s + 0];
    const float* b0 = (const float*)d_in[1 + 6 * s + 1];
    const float* W1 = (const float*)d_in[1 + 6 * s + 2];
    const float* b1 = (const float*)d_in[1 + 6 * s + 3];
    const float* W2 = (const float*)d_in[1 + 6 * s + 4];
    const float* b2 = (const float*)d_in[1 + 6 * s + 5];
    const float* Wa = (const float*)d_in[13 + 6 * s + 0];
    const float* ba = (const float*)d_in[13 + 6 * s + 1];
    const float* Wb = (const float*)d_in[13 + 6 * s + 2];
    const float* bb = (const float*)d_in[13 + 6 * s + 3];
    const float* Wc = (const float*)d_in[13 + 6 * s + 4];
    const float* bc = (const float*)d_in[13 + 6 * s + 5];

    pq_kernel  <<<NODES, 192, 0, stream>>>(xcur, W0, b0, P, Q);
    edge_kernel<<<NODES, 320, 0, stream>>>(xcur, P, Q, W0 + 64 * E0,
                                           W1, b1, W2, b2, agg);
    node_kernel<<<NODES / 16, 256, 0, stream>>>(agg, xcur, Wa, ba, Wb, bb, Wc, bc,
                                                xnext);
    xcur = xnext; xnext = xb1;
  }
  tanh_kernel<<<(NODES * 3 + 255) / 256, 256, 0, stream>>>(xcur, (float*)d_out);
}